// DisturbanceRegressionLoss2Heads_41446434406841
// MI455X (gfx1250) — compile-verified
//
#include <hip/hip_runtime.h>

typedef float v2f __attribute__((ext_vector_type(2)));
typedef float v8f __attribute__((ext_vector_type(8)));

#define BN 8
#define YN 30
#define HN 256
#define WN 256
#define PLANE (HN * WN)            /* 65536 floats per (b,y) plane   */
#define HEAD_STRIDE (YN * PLANE)   /* stride between head 0 / head 1 */
#define BATCH_STRIDE (2 * HEAD_STRIDE)
#define NPIX (BN * PLANE)          /* 524288 pixels                  */
#define NPAIR (NPIX / 2)           /* 262144 threads, 2 pixels each  */
#define NTOT_INV (1.0f / (float)(BN * YN * PLANE)) /* 1/15728640     */

// ---------------------------------------------------------------------------
// Per-pixel breakpoint + two masked OLS fits, all in registers.
// v[0..29] = out0 time series for this pixel.
// Returns d and clipped slope / clipped intercept for both segments.
// ---------------------------------------------------------------------------
__device__ __forceinline__ void fit30(const float v[YN], int& d,
                                      float& sbc, float& ib,
                                      float& sac, float& ia)
{
    // diff[0] = -7 (sentinel), diff[1] = 0, diff[29] = 0 can never beat -7
    // under strict '<' (first-occurrence argmin), so only y = 2..28 scanned.
    float dmin = -7.0f;
    d = 0;
#pragma unroll
    for (int y = 2; y <= 28; ++y) {
        float df = v[y] - v[y - 1];
        if (df < dmin) { dmin = df; d = y; }
    }

    // Data-dependent sums: totals and prefix (t < d).
    float Sy = 0.0f, Sty = 0.0f, Syb = 0.0f, Styb = 0.0f;
#pragma unroll
    for (int y = 0; y < YN; ++y) {
        float fy = (float)y;
        Sy  += v[y];
        Sty  = fmaf(fy, v[y], Sty);
        if (y < d) { Syb += v[y]; Styb = fmaf(fy, v[y], Styb); }
    }

    // Segment "before": x = t, n = d.   mean_x = (Σ_{t<d} t)/max(n,1)
    float nb   = (float)d;
    float nbs  = fmaxf(nb, 1.0f);
    float mxb  = (0.5f * nb * (nb - 1.0f)) / nbs;
    float myb  = Syb / nbs;
    float covb = Styb - mxb * Syb;                  // Σ(x-mx)(y-my)mask
    float varb = nb * (nb * nb - 1.0f) * (1.0f / 12.0f);
    float slb  = (varb > 0.0f) ? covb / fmaxf(varb, 1.0f) : 0.0f;
    ib  = fminf(fmaxf(myb - slb * mxb, 0.0f), 100.0f);
    sbc = fminf(fmaxf(slb, 0.0f), 2.0f);

    // Segment "after": x = t - d, n = 30 - d  (>= 2 always).
    float na   = (float)YN - nb;
    float Sya  = Sy - Syb;
    float Sxya = (Sty - Styb) - nb * Sya;           // Σ (t-d) y over t>=d
    float mxa  = 0.5f * (na - 1.0f);
    float mya  = Sya / na;
    float cova = Sxya - mxa * Sya;
    float vara = na * (na * na - 1.0f) * (1.0f / 12.0f);
    float sla  = (vara > 0.0f) ? cova / fmaxf(vara, 1.0f) : 0.0f;
    ia  = fminf(fmaxf(mya - sla * mxa, 0.0f), 100.0f);
    sac = fminf(fmaxf(sla, 0.0f), 2.0f);
}

__global__ void zero_scalar_kernel(float* __restrict__ o) { o[0] = 0.0f; }

// ---------------------------------------------------------------------------
// One thread = 2 adjacent-w pixels -> every global access is an 8B b64 load
// (256B per wave32 per instruction). Block partials are reduced exactly in
// f32 via V_WMMA_F32_16X16X4_F32 against an all-ones B matrix.
// ---------------------------------------------------------------------------
__global__ __launch_bounds__(256)
void disturbance_loss_kernel(const float* __restrict__ out,
                             float* __restrict__ loss)
{
    const int tid    = threadIdx.x;
    const int pair   = blockIdx.x * 256 + tid;      // 0 .. NPAIR-1
    const int pixel0 = pair << 1;
    const int b      = pixel0 >> 16;                // / PLANE
    const int hw     = pixel0 & (PLANE - 1);

    const float* base0 = out + (size_t)b * BATCH_STRIDE + hw; // head 0
    const float* base1 = base0 + HEAD_STRIDE;                 // head 1

    // ---- stream out0 for both pixels (coalesced b64 loads) ----
    float a0[YN], a1[YN];
#pragma unroll
    for (int y = 0; y < YN; ++y) {
        v2f t = *(const v2f*)(base0 + (size_t)y * PLANE);
        a0[y] = t.x;
        a1[y] = t.y;
    }

    // ---- prefetch out1 planes (global_prefetch) while fit math runs ----
#pragma unroll
    for (int y = 0; y < YN; ++y)
        __builtin_prefetch(base1 + (size_t)y * PLANE, 0, 1);

    int   d0, d1;
    float s0b, i0b, s0a, i0a;
    float s1b, i1b, s1a, i1a;
    fit30(a0, d0, s0b, i0b, s0a, i0a);
    fit30(a1, d1, s1b, i1b, s1a, i1a);

    // ---- stream out1, accumulate squared error ----
    float acc = 0.0f;
#pragma unroll
    for (int y = 0; y < YN; ++y) {
        v2f t = *(const v2f*)(base1 + (size_t)y * PLANE);
        float f0 = (y < d0) ? fmaf(s0b, (float)y, i0b)
                            : fmaf(s0a, (float)(y - d0), i0a);
        float f1 = (y < d1) ? fmaf(s1b, (float)y, i1b)
                            : fmaf(s1a, (float)(y - d1), i1a);
        float e0 = f0 - t.x;
        float e1 = f1 - t.y;
        acc = fmaf(e0, e0, acc);
        acc = fmaf(e1, e1, acc);
    }

    // ---- block reduction: 256 partials -> 1, exact f32 via WMMA ----
    __shared__ float sdata[256];
    sdata[tid] = acc * NTOT_INV;
    __syncthreads();

    if (tid < 32) {                      // wave 0, EXEC all ones
        const int lane = tid;
        const int row  = lane & 15;           // A-matrix row M
        const int koff = (lane >> 4) << 1;    // lanes 0-15: K0/K1, 16-31: K2/K3

        v8f c = {0.f, 0.f, 0.f, 0.f, 0.f, 0.f, 0.f, 0.f};
        v2f bones = {1.0f, 1.0f};             // B = ones(4x16)
#pragma unroll
        for (int chunk = 0; chunk < 4; ++chunk) {
            v2f A;
            A.x = sdata[chunk * 64 + row * 4 + koff];
            A.y = sdata[chunk * 64 + row * 4 + koff + 1];
            // D = A(16x4) * ones(4x16) + C : every column n holds the row sums
            c = __builtin_amdgcn_wmma_f32_16x16x4_f32(
                    false, A, false, bones, (short)0, c, false, false);
        }
        // lane holds D[m][n=lane%16] for m = 0..7 (lanes<16) or 8..15 in its
        // 8 VGPRs; all columns identical, so local sum + xor-16 = block total.
        float s = c[0] + c[1] + c[2] + c[3] + c[4] + c[5] + c[6] + c[7];
        s += __shfl_xor(s, 16, 32);
        if (lane == 0)
            atomicAdd(loss, s);
    }
}

extern "C" void kernel_launch(void* const* d_in, const int* in_sizes, int n_in,
                              void* d_out, int out_size, void* d_ws, size_t ws_size,
                              hipStream_t stream)
{
    const float* out = (const float*)d_in[0];   // (8,2,30,256,256) f32
    // d_in[1] (target) is unused by the reference loss.
    float* loss = (float*)d_out;                // scalar f32

    zero_scalar_kernel<<<1, 1, 0, stream>>>(loss);
    disturbance_loss_kernel<<<NPAIR / 256, 256, 0, stream>>>(out, loss);
}